// Sparsemax_46076409151920
// MI455X (gfx1250) — compile-verified
//
#include <hip/hip_runtime.h>
#include <hip/hip_bf16.h>
#include <cstdint>

typedef __attribute__((ext_vector_type(4))) unsigned int u32x4;
typedef __attribute__((ext_vector_type(8))) int          i32x8;
typedef __attribute__((ext_vector_type(4))) int          i32x4;
typedef __attribute__((ext_vector_type(4))) float        f32x4;

#define NLOG    4096
#define THREADS 256
#define MAXIT   32

__global__ __launch_bounds__(THREADS)
void sparsemax_kernel(const float* __restrict__ x, float* __restrict__ out) {
    __shared__ float row[NLOG];     // 16 KB staged row (TDM destination)
    __shared__ float rmax[8];
    __shared__ float rsum[2][8];
    __shared__ float rcnt[2][8];

    const int tid  = threadIdx.x;
    const int lane = tid & 31;
    const int wid  = tid >> 5;
    const long long r = blockIdx.x;
    const float* src = x + r * (long long)NLOG;

    // ---- Stage row into LDS with one Tensor Data Mover op (wave 0 only) ----
    if (wid == 0) {
        unsigned long long ga = (unsigned long long)(uintptr_t)src;
        unsigned int lds_off  = (unsigned int)(uintptr_t)(void*)&row[0];

        u32x4 g0;
        g0.x = 1u;                                     // count=1 (valid D#)
        g0.y = lds_off;                                // lds_addr (bytes)
        g0.z = (unsigned int)ga;                       // global_addr[31:0]
        g0.w = (unsigned int)(ga >> 32) | (2u << 30);  // global_addr[56:32] | type=2

        i32x8 g1;
        g1[0] = 0x00020000;                            // data_size=2 (4 bytes)
        g1[1] = (int)((NLOG & 0xFFFFu) << 16);         // tensor_dim0[15:0]=4096
        g1[2] = (int)((NLOG >> 16) | (1u << 16));      // tensor_dim0[31:16] | tensor_dim1=1
        g1[3] = (int)((NLOG & 0xFFFFu) << 16);         // tensor_dim1[31:16]=0 | tile_dim0=4096
        g1[4] = 1;                                     // tile_dim1=1, tile_dim2=0
        g1[5] = NLOG;                                  // tensor_dim0_stride[31:0]
        g1[6] = 0;                                     // stride0 hi | stride1 lo
        g1[7] = 0;                                     // stride1 hi

        i32x4 z4 = {0, 0, 0, 0};                       // groups 2/3: dims beyond 2D unused
        i32x8 z8 = {0, 0, 0, 0, 0, 0, 0, 0};
#if __has_builtin(__builtin_amdgcn_tensor_load_to_lds)
        __builtin_amdgcn_tensor_load_to_lds(g0, g1, z4, z4, z8, 0);
        __builtin_amdgcn_s_wait_tensorcnt(0);
#else
        for (int i = tid; i < NLOG; i += 32) row[i] = src[i];
#endif
    }
    __syncthreads();

    // ---- Pull 16 elements/thread from LDS into registers (f32x4 x4) ----
    f32x4 v[4];
#pragma unroll
    for (int j = 0; j < 4; ++j)
        v[j] = *(const f32x4*)&row[tid * 4 + j * 1024];

    // ---- Row max (wave32 shuffle reduce + 8-wave LDS fold) ----
    float m = -3.402823466e38f;
#pragma unroll
    for (int j = 0; j < 4; ++j) {
        m = fmaxf(m, fmaxf(fmaxf(v[j].x, v[j].y), fmaxf(v[j].z, v[j].w)));
    }
#pragma unroll
    for (int d = 16; d >= 1; d >>= 1) m = fmaxf(m, __shfl_xor(m, d, 32));
    if (lane == 0) rmax[wid] = m;
    __syncthreads();
    float zmax = rmax[0];
#pragma unroll
    for (int i = 1; i < 8; ++i) zmax = fmaxf(zmax, rmax[i]);

    // shift: w = z - max  (sparsemax is shift-invariant; matches reference numerics)
#pragma unroll
    for (int j = 0; j < 4; ++j) {
        v[j].x -= zmax; v[j].y -= zmax; v[j].z -= zmax; v[j].w -= zmax;
    }

    // ---- Exact tau via fixed-point threshold iteration (no sort) ----
    // tau_{t+1} = (sum_{w>tau_t} w - 1) / |{w>tau_t}| ; tau_0 = -inf -> full-sum init.
    // All threads compute bit-identical tau each step (same LDS partials, same
    // fold order), so the convergence break is a uniform branch.
    float tau = -3.402823466e38f;
    for (int it = 0; it < MAXIT; ++it) {
        float s = 0.0f, c = 0.0f;
#pragma unroll
        for (int j = 0; j < 4; ++j) {
            const f32x4 w = v[j];
            if (w.x > tau) { s += w.x; c += 1.0f; }
            if (w.y > tau) { s += w.y; c += 1.0f; }
            if (w.z > tau) { s += w.z; c += 1.0f; }
            if (w.w > tau) { s += w.w; c += 1.0f; }
        }
#pragma unroll
        for (int d = 16; d >= 1; d >>= 1) {
            s += __shfl_xor(s, d, 32);
            c += __shfl_xor(c, d, 32);
        }
        const int b = it & 1;                 // double-buffered partials: 1 barrier/iter
        if (lane == 0) { rsum[b][wid] = s; rcnt[b][wid] = c; }
        __syncthreads();
        float S = 0.0f, C = 0.0f;
#pragma unroll
        for (int i = 0; i < 8; ++i) { S += rsum[b][i]; C += rcnt[b][i]; }
        const float tnew = (S - 1.0f) / C;    // C >= 1 always (w_max = 0 > tau)
        if (tnew == tau) break;               // uniform: exact fixed point reached
        tau = tnew;
    }

    // ---- p = max(w - tau, 0), coalesced non-temporal f32x4 stores ----
    float* dst = out + r * (long long)NLOG;
#pragma unroll
    for (int j = 0; j < 4; ++j) {
        f32x4 p;
        p.x = fmaxf(v[j].x - tau, 0.0f);
        p.y = fmaxf(v[j].y - tau, 0.0f);
        p.z = fmaxf(v[j].z - tau, 0.0f);
        p.w = fmaxf(v[j].w - tau, 0.0f);
        __builtin_nontemporal_store(p, (f32x4*)&dst[tid * 4 + j * 1024]);
    }
}

extern "C" void kernel_launch(void* const* d_in, const int* in_sizes, int n_in,
                              void* d_out, int out_size, void* d_ws, size_t ws_size,
                              hipStream_t stream) {
    const float* x = (const float*)d_in[0];
    float* out = (float*)d_out;
    const int rows = in_sizes[0] / NLOG;   // 16384
    sparsemax_kernel<<<rows, THREADS, 0, stream>>>(x, out);
}